// MaSAd_75453985457151
// MI455X (gfx1250) — compile-verified
//
#include <hip/hip_runtime.h>
#include <hip/hip_bf16.h>

typedef __attribute__((ext_vector_type(2))) float v2f;
typedef __attribute__((ext_vector_type(8))) float v8f;
typedef __attribute__((ext_vector_type(4))) int   v4i;

__device__ __forceinline__ v8f wmma4(v2f a, v2f b, v8f c) {
    // V_WMMA_F32_16X16X4_F32 : D = A(16x4) * B(4x16) + C(16x16)
    return __builtin_amdgcn_wmma_f32_16x16x4_f32(
        /*neg_a=*/false, a, /*neg_b=*/false, b,
        /*c_mod=*/(short)0, c, /*reuse_a=*/false, /*reuse_b=*/false);
}

// ---- CDNA5 async global->LDS staging (ASYNCcnt path), with safe fallback ----
#if defined(__has_builtin)
#  if __has_builtin(__builtin_amdgcn_global_load_async_to_lds_b128)
#    define USE_ASYNC_LDS 1
#  endif
#endif
#ifndef USE_ASYNC_LDS
#  define USE_ASYNC_LDS 0
#endif

#if defined(__has_builtin)
#  if __has_builtin(__builtin_amdgcn_s_wait_asynccnt)
#    define HAVE_WAIT_ASYNC_BUILTIN 1
#  endif
#endif
#ifndef HAVE_WAIT_ASYNC_BUILTIN
#  define HAVE_WAIT_ASYNC_BUILTIN 0
#endif

__device__ __forceinline__ void stage16(const void* g, void* l) {
#if USE_ASYNC_LDS
    typedef __attribute__((address_space(1))) v4i GV;   // global v4i*
    typedef __attribute__((address_space(3))) v4i LV;   // LDS v4i*
    __builtin_amdgcn_global_load_async_to_lds_b128(
        (GV*)g, (LV*)l, /*offset=*/0, /*cpol=*/0);
#else
    *reinterpret_cast<float4*>(l) = *reinterpret_cast<const float4*>(g);
#endif
}

__device__ __forceinline__ void stage_join() {
#if USE_ASYNC_LDS
#  if HAVE_WAIT_ASYNC_BUILTIN
    __builtin_amdgcn_s_wait_asynccnt(0);
#  else
    asm volatile("s_wait_asynccnt 0x0" ::: "memory");
#  endif
#endif
    __syncthreads();
}

// ---------------------------------------------------------------------------
// Kernel 1: fused QKV projection.  q = x@Wq^T+bq, k = (x@Wk^T+bk)*scale,
// v = x@Wv^T+bv.  M=8192 rows (r*512+p), N=C=512, K=512.
// grid (512 Mtiles, 4 Ngroups), block 256 (8 waves); wave w covers N-tile
// blockIdx.y*8+w; X tile staged in LDS, A-frag shared across the 3 GEMMs.
// ---------------------------------------------------------------------------
__global__ __launch_bounds__(256) void qkv_kernel(
    const float* __restrict__ x,
    const float* __restrict__ Wq, const float* __restrict__ bq,
    const float* __restrict__ Wk, const float* __restrict__ bk,
    const float* __restrict__ Wv, const float* __restrict__ bv,
    float* __restrict__ q, float* __restrict__ k, float* __restrict__ v)
{
    __shared__ float Xs[16 * 516];                 // padded stride: no bank conflicts
    const int tid   = threadIdx.x;
    const int mbase = blockIdx.x * 16;

    for (int i = tid; i < 16 * 128; i += 256) {
        int row = i >> 7, c4 = (i & 127) << 2;
        stage16(x + (size_t)(mbase + row) * 512 + c4, &Xs[row * 516 + c4]);
    }
    stage_join();

    const int lane = tid & 31, wid = tid >> 5;
    const int hi = lane >> 4, ln = lane & 15;
    const int n0 = (blockIdx.y * 8 + wid) * 16;

    v8f accq = {}, acck = {}, accv = {};
    const float* As = &Xs[ln * 516 + 2 * hi];
    const float* Bq = Wq + (size_t)(n0 + ln) * 512 + 2 * hi;
    const float* Bk = Wk + (size_t)(n0 + ln) * 512 + 2 * hi;
    const float* Bv = Wv + (size_t)(n0 + ln) * 512 + 2 * hi;

    #pragma unroll 4
    for (int k0 = 0; k0 < 512; k0 += 4) {
        v2f a  = *reinterpret_cast<const v2f*>(As + k0);   // ds_load_b64
        v2f b0 = *reinterpret_cast<const v2f*>(Bq + k0);
        accq = wmma4(a, b0, accq);
        v2f b1 = *reinterpret_cast<const v2f*>(Bk + k0);
        acck = wmma4(a, b1, acck);
        v2f b2 = *reinterpret_cast<const v2f*>(Bv + k0);
        accv = wmma4(a, b2, accv);
    }

    const float scaling = 0.17677669529663687f;    // 32^-0.5
    const float bqv = bq[n0 + ln], bkv = bk[n0 + ln], bvv = bv[n0 + ln];
    #pragma unroll
    for (int vr = 0; vr < 8; ++vr) {
        int row = mbase + vr + 8 * hi;
        size_t o = (size_t)row * 512 + n0 + ln;
        q[o] = accq[vr] + bqv;
        k[o] = (acck[vr] + bkv) * scaling;
        v[o] = accv[vr] + bvv;
    }
}

// ---------------------------------------------------------------------------
// Kernel 2: depthwise 5x5 conv (lepe), NHWC, pad 2, spatial (16, 512), C=512.
// ---------------------------------------------------------------------------
__global__ __launch_bounds__(256) void lepe_kernel(
    const float* __restrict__ v, const float* __restrict__ w,
    const float* __restrict__ b, float* __restrict__ out)
{
    int idx = blockIdx.x * 256 + threadIdx.x;      // 16*512*512 total, c fastest
    int c = idx & 511;
    int p = (idx >> 9) & 511;
    int r = idx >> 18;
    float acc = b[c];
    #pragma unroll
    for (int dy = 0; dy < 5; ++dy) {
        int rr = r + dy - 2;
        if ((unsigned)rr >= 16u) continue;
        #pragma unroll
        for (int dx = 0; dx < 5; ++dx) {
            int pp = p + dx - 2;
            if ((unsigned)pp >= 512u) continue;
            acc += v[((size_t)rr * 512 + pp) * 512 + c] * w[(dy * 5 + dx) * 512 + c];
        }
    }
    out[idx] = acc;
}

// ---------------------------------------------------------------------------
// Kernels 3/4: decomposed attention, flash-style online softmax.
// Pair (r,n): Q,K from (row=r, head=n).
//   pass 1: mask_h[r], V = v(row=n, head=r)  -> v1'[r][n][p][d]
//   pass 2: mask_w[n], V = v1'[r][n]          -> attn[r][p][n*32+d]
// Block = 128 threads (4 waves), grid (256 pairs, 8 rowblock-groups).
// K,V tiles (512x32 each) staged fully in LDS via async global->LDS;
// P bounced C->A layout via wave-private LDS.
// ---------------------------------------------------------------------------
__global__ __launch_bounds__(128) void attn_kernel(
    const float* __restrict__ q, const float* __restrict__ k,
    const float* __restrict__ vsrc, const float* __restrict__ mask,
    float* __restrict__ out, int pass)
{
    __shared__ float Ks[512 * 36];                 // [u][d], stride 36 (conflict-free)
    __shared__ float Vs[512 * 36];
    __shared__ float Pb[4][16 * 18];               // per-wave P bounce, stride 18

    const int pair = blockIdx.x;                   // 0..255
    const int r = pair >> 4, n = pair & 15;
    const int tid = threadIdx.x, lane = tid & 31, wid = tid >> 5;
    const int hi = lane >> 4, ln = lane & 15;

    const float* kbase = k + (size_t)r * 512 * 512 + n * 32;
    const float* vbase;
    size_t vstride;
    if (pass == 1) { vbase = vsrc + (size_t)n * 512 * 512 + r * 32; vstride = 512; }
    else           { vbase = vsrc + (size_t)(r * 16 + n) * 512 * 32; vstride = 32; }
    const float* mbase = mask + (size_t)(pass == 1 ? r : n) * 512 * 512;

    for (int i = tid; i < 512 * 8; i += 128) {
        int u = i >> 3, d4 = (i & 7) << 2;
        stage16(kbase + (size_t)u * 512 + d4, &Ks[u * 36 + d4]);
        stage16(vbase + (size_t)u * vstride + d4, &Vs[u * 36 + d4]);
    }
    stage_join();

    const int m0 = (blockIdx.y * 4 + wid) * 16;    // this wave's 16 query rows

    // preload Q A-fragments (16x32 tile -> 8 k-steps)
    const float* qrow = q + (size_t)(r * 512 + m0 + ln) * 512 + n * 32 + 2 * hi;
    v2f aq[8];
    #pragma unroll
    for (int kk = 0; kk < 8; ++kk)
        aq[kk] = *reinterpret_cast<const v2f*>(qrow + kk * 4);

    v8f o0 = {}, o1 = {};                          // O tile 16 x 32 (two 16-col chunks)
    float rmax[8], rsum[8];
    #pragma unroll
    for (int i = 0; i < 8; ++i) { rmax[i] = -3.0e38f; rsum[i] = 0.0f; }

    float* pb = Pb[wid];
    const float* maskrow = mbase + (size_t)m0 * 512;

    for (int jb = 0; jb < 32; ++jb) {
        const int u0 = jb * 16;

        // prefetch next mask block (streams once from HBM) -> global_prefetch
        __builtin_prefetch(maskrow + (size_t)(8 * hi) * 512 + u0 + 16 + ln, 0, 0);

        // S tile = Q K^T  (16x16, K-dim 32 via 8 WMMA)
        v8f s = {};
        const float* kb = &Ks[(u0 + ln) * 36 + 2 * hi];
        #pragma unroll
        for (int kk = 0; kk < 8; ++kk) {
            v2f b = *reinterpret_cast<const v2f*>(kb + kk * 4);
            s = wmma4(aq[kk], b, s);
        }

        // mask add + online softmax update (stats kept in C layout)
        #pragma unroll
        for (int vr = 0; vr < 8; ++vr) {
            float sv = s[vr] + maskrow[(size_t)(vr + 8 * hi) * 512 + u0 + ln];
            float x = sv;                           // row max over 16 lanes
            x = fmaxf(x, __shfl_xor(x, 1));
            x = fmaxf(x, __shfl_xor(x, 2));
            x = fmaxf(x, __shfl_xor(x, 4));
            x = fmaxf(x, __shfl_xor(x, 8));
            float newm = fmaxf(rmax[vr], x);
            float corr = __expf(rmax[vr] - newm);
            rmax[vr] = newm;
            float p = __expf(sv - newm);
            float ps = p;                           // row sum of exp
            ps += __shfl_xor(ps, 1);
            ps += __shfl_xor(ps, 2);
            ps += __shfl_xor(ps, 4);
            ps += __shfl_xor(ps, 8);
            rsum[vr] = rsum[vr] * corr + ps;
            o0[vr] *= corr;
            o1[vr] *= corr;
            s[vr] = p;
        }

        // bounce P (C layout) -> LDS -> A layout (wave-private, DS in-order)
        #pragma unroll
        for (int vr = 0; vr < 8; ++vr)
            pb[(vr + 8 * hi) * 18 + ln] = s[vr];

        // O += P * V  (K-dim 16 -> 4 WMMA per 16-col chunk)
        #pragma unroll
        for (int kk = 0; kk < 4; ++kk) {
            v2f ap = *reinterpret_cast<const v2f*>(&pb[ln * 18 + kk * 4 + 2 * hi]);
            int ua = (u0 + kk * 4 + 2 * hi) * 36;
            v2f b0; b0.x = Vs[ua + ln];      b0.y = Vs[ua + 36 + ln];
            o0 = wmma4(ap, b0, o0);
            v2f b1; b1.x = Vs[ua + 16 + ln]; b1.y = Vs[ua + 36 + 16 + ln];
            o1 = wmma4(ap, b1, o1);
        }
    }

    #pragma unroll
    for (int vr = 0; vr < 8; ++vr) {
        float inv = 1.0f / rsum[vr];
        o0[vr] *= inv;
        o1[vr] *= inv;
    }

    if (pass == 1) {
        float* ob = out + (size_t)(r * 16 + n) * 512 * 32;      // v1'[r][n][p][d]
        #pragma unroll
        for (int vr = 0; vr < 8; ++vr) {
            int p_ = m0 + vr + 8 * hi;
            ob[(size_t)p_ * 32 + ln]      = o0[vr];
            ob[(size_t)p_ * 32 + 16 + ln] = o1[vr];
        }
    } else {
        float* ob = out + (size_t)r * 512 * 512 + n * 32;       // attn[r][p][n*32+d]
        #pragma unroll
        for (int vr = 0; vr < 8; ++vr) {
            int p_ = m0 + vr + 8 * hi;
            ob[(size_t)p_ * 512 + ln]      = o0[vr];
            ob[(size_t)p_ * 512 + 16 + ln] = o1[vr];
        }
    }
}

// ---------------------------------------------------------------------------
// Kernel 5: output projection  out = (attn + lepe) @ Wo^T + bo
// (synchronous staging: the add happens on the way into LDS)
// ---------------------------------------------------------------------------
__global__ __launch_bounds__(256) void proj_kernel(
    const float* __restrict__ attn, const float* __restrict__ lepe,
    const float* __restrict__ Wo, const float* __restrict__ bo,
    float* __restrict__ out)
{
    __shared__ float Xs[16 * 516];
    const int tid   = threadIdx.x;
    const int mbase = blockIdx.x * 16;

    for (int i = tid; i < 16 * 128; i += 256) {
        int row = i >> 7, c4 = (i & 127) << 2;
        size_t off = (size_t)(mbase + row) * 512 + c4;
        float4 a4 = *reinterpret_cast<const float4*>(attn + off);
        float4 l4 = *reinterpret_cast<const float4*>(lepe + off);
        a4.x += l4.x; a4.y += l4.y; a4.z += l4.z; a4.w += l4.w;
        *reinterpret_cast<float4*>(&Xs[row * 516 + c4]) = a4;
    }
    __syncthreads();

    const int lane = tid & 31, wid = tid >> 5;
    const int hi = lane >> 4, ln = lane & 15;
    const int n0 = (blockIdx.y * 8 + wid) * 16;

    v8f acc = {};
    const float* As = &Xs[ln * 516 + 2 * hi];
    const float* Bo = Wo + (size_t)(n0 + ln) * 512 + 2 * hi;

    #pragma unroll 8
    for (int k0 = 0; k0 < 512; k0 += 4) {
        v2f a = *reinterpret_cast<const v2f*>(As + k0);
        v2f b = *reinterpret_cast<const v2f*>(Bo + k0);
        acc = wmma4(a, b, acc);
    }

    const float bov = bo[n0 + ln];
    #pragma unroll
    for (int vr = 0; vr < 8; ++vr) {
        int row = mbase + vr + 8 * hi;
        out[(size_t)row * 512 + n0 + ln] = acc[vr] + bov;
    }
}

// ---------------------------------------------------------------------------
extern "C" void kernel_launch(void* const* d_in, const int* in_sizes, int n_in,
                              void* d_out, int out_size, void* d_ws, size_t ws_size,
                              hipStream_t stream)
{
    const float* x      = (const float*)d_in[0];
    const float* mask_h = (const float*)d_in[1];
    const float* mask_w = (const float*)d_in[2];
    const float* Wq     = (const float*)d_in[3];
    const float* bq     = (const float*)d_in[4];
    const float* Wk     = (const float*)d_in[5];
    const float* bk     = (const float*)d_in[6];
    const float* Wv     = (const float*)d_in[7];
    const float* bv     = (const float*)d_in[8];
    const float* lepe_w = (const float*)d_in[9];
    const float* lepe_b = (const float*)d_in[10];
    const float* Wo     = (const float*)d_in[11];
    const float* bo     = (const float*)d_in[12];
    (void)in_sizes; (void)n_in; (void)out_size; (void)ws_size;

    const size_t NELT = (size_t)8192 * 512;        // 4,194,304 floats per tensor
    float* q    = (float*)d_ws;
    float* kk   = q    + NELT;
    float* vv   = kk   + NELT;
    float* lepe = vv   + NELT;
    float* v1   = lepe + NELT;
    float* attn = v1   + NELT;

    qkv_kernel<<<dim3(512, 4), 256, 0, stream>>>(x, Wq, bq, Wk, bk, Wv, bv, q, kk, vv);
    lepe_kernel<<<16384, 256, 0, stream>>>(vv, lepe_w, lepe_b, lepe);
    attn_kernel<<<dim3(256, 8), 128, 0, stream>>>(q, kk, vv, mask_h, v1, 1);
    attn_kernel<<<dim3(256, 8), 128, 0, stream>>>(q, kk, v1, mask_w, attn, 2);
    proj_kernel<<<dim3(512, 4), 256, 0, stream>>>(attn, lepe, Wo, bo, (float*)d_out);
}